// Fusion_Block_GLAFF_15728170238234
// MI455X (gfx1250) — compile-verified
//
#include <hip/hip_runtime.h>
#include <math.h>
#include <stdint.h>

#define B_  32
#define L_  512
#define D_  32
#define K_  6
#define O_  192
#define DM_ 512
#define KP_ 8   // K padded to 8 for two 16x16x4 f32 WMMA k-steps

typedef __attribute__((ext_vector_type(2))) float v2f;
typedef __attribute__((ext_vector_type(8))) float v8f;
typedef __attribute__((ext_vector_type(4))) unsigned int u32x4;
typedef __attribute__((ext_vector_type(8))) unsigned int u32x8;

// ---------------------------------------------------------------------------
// Kernel 1: per-column bitonic sort (length 512) -> median (torch lower) and
// interquantile range (q=0.75 / 0.25, linear interpolation like jnp.quantile).
// ---------------------------------------------------------------------------
__global__ void __launch_bounds__(512)
sort_stats_kernel(const float* __restrict__ in, int inner,
                  float* __restrict__ med, float* __restrict__ stdv) {
    __shared__ float s[L_];
    const int c = blockIdx.x;
    const size_t base = (size_t)(c / inner) * (L_ * inner) + (c % inner);
    const int i = threadIdx.x;
    s[i] = in[base + (size_t)i * inner];
    __syncthreads();
    for (int k = 2; k <= L_; k <<= 1) {
        for (int j = k >> 1; j > 0; j >>= 1) {
            int ixj = i ^ j;
            if (ixj > i) {
                float a = s[i], b = s[ixj];
                bool up = ((i & k) == 0);
                if (up ? (a > b) : (a < b)) { s[i] = b; s[ixj] = a; }
            }
            __syncthreads();
        }
    }
    if (i == 0) {
        float m  = s[255];                               // lower median, n=512
        float qh = 0.75f * s[383] + 0.25f * s[384];      // q=0.75 @ 383.25
        float ql = 0.25f * s[127] + 0.75f * s[128];      // q=0.25 @ 127.75
        med[c]  = m;
        stdv[c] = qh - ql + 1e-6f;
    }
}

// ---------------------------------------------------------------------------
// Kernel 2: x_date_map (K-padded, [b][d][l][8]) + error = x - mean_k(map),
// error stored [b][d][l] (contiguous l for the MLP GEMM A-operand loads).
// ---------------------------------------------------------------------------
__global__ void map_x_kernel(const float* __restrict__ x,
                             const float* __restrict__ x_date,
                             const float* __restrict__ med_t, const float* __restrict__ std_t,
                             const float* __restrict__ med_m, const float* __restrict__ std_m,
                             float* __restrict__ xmap, float* __restrict__ err) {
    int idx = blockIdx.x * blockDim.x + threadIdx.x;
    if (idx >= B_ * L_ * D_) return;
    int d = idx % D_;
    int l = (idx / D_) % L_;
    int b = idx / (L_ * D_);
    int bd = b * D_ + d;
    float mt = med_t[bd], st = std_t[bd];
    const float* xd = x_date + (size_t)idx * K_;
    float* outr = xmap + ((size_t)bd * L_ + l) * KP_;
    float acc = 0.f;
#pragma unroll
    for (int k = 0; k < K_; ++k) {
        float v = (xd[k] - med_m[bd * K_ + k]) / std_m[bd * K_ + k] * st + mt;
        outr[k] = v;
        acc += v;
    }
    outr[6] = 0.f; outr[7] = 0.f;
    err[(size_t)bd * L_ + l] = x[idx] - acc * (1.f / 6.f);
}

// ---------------------------------------------------------------------------
// Kernel 3: y_date_map (K-padded, [b][d][o][8]) + ymean [b][o][d].
// ---------------------------------------------------------------------------
__global__ void map_y_kernel(const float* __restrict__ y_date,
                             const float* __restrict__ med_t, const float* __restrict__ std_t,
                             const float* __restrict__ med_m, const float* __restrict__ std_m,
                             float* __restrict__ ymap, float* __restrict__ ymean) {
    int idx = blockIdx.x * blockDim.x + threadIdx.x;
    if (idx >= B_ * O_ * D_) return;
    int d = idx % D_;
    int o = (idx / D_) % O_;
    int b = idx / (O_ * D_);
    int bd = b * D_ + d;
    float mt = med_t[bd], st = std_t[bd];
    const float* yd = y_date + (size_t)idx * K_;
    float* outr = ymap + ((size_t)bd * O_ + o) * KP_;
    float acc = 0.f;
#pragma unroll
    for (int k = 0; k < K_; ++k) {
        float v = (yd[k] - med_m[bd * K_ + k]) / std_m[bd * K_ + k] * st + mt;
        outr[k] = v;
        acc += v;
    }
    outr[6] = 0.f; outr[7] = 0.f;
    ymean[idx] = acc * (1.f / 6.f);
}

// ---------------------------------------------------------------------------
// Kernel 4: fused attention per (b,d): WMMA 16x16 score tiles (two f32
// 16x16x4 k-steps over padded K=8), online softmax over L, and V = x^T A.
// ---------------------------------------------------------------------------
__global__ void __launch_bounds__(384)
attn_kernel(const float* __restrict__ x, const float* __restrict__ xmap,
            const float* __restrict__ ymap, float* __restrict__ V) {
    __shared__ float xs[L_];
    const int bd = blockIdx.x;
    const int b = bd / D_, d = bd % D_;
    for (int i = threadIdx.x; i < L_; i += 384)
        xs[i] = x[((size_t)b * L_ + i) * D_ + d];
    __syncthreads();

    const int wid  = threadIdx.x >> 5;   // 0..11 -> O-tile
    const int lane = threadIdx.x & 31;
    const int half = lane >> 4;          // 0: k{0,1}/rows0-7, 1: k{2,3}/rows8-15
    const int lr   = lane & 15;
    const int o    = wid * 16 + lr;

    const float* yrow = ymap + ((size_t)bd * O_ + o) * KP_;
    v2f b0 = *(const v2f*)(yrow + half * 2);       // k-step 0: k {0,1} / {2,3}
    v2f b1 = *(const v2f*)(yrow + 4 + half * 2);   // k-step 1: k {4,5} / {6,7}=0

    const float inv_sqrt_k = 0.4082482904638630f;  // 1/sqrt(6)
    float m = -3.0e38f, ssum = 0.f, vacc = 0.f;
    const float* abase = xmap + (size_t)bd * L_ * KP_;

    for (int lt = 0; lt < L_ / 16; ++lt) {
        const int l0 = lt * 16;
        const float* arow = abase + (size_t)(l0 + lr) * KP_ + half * 2;
        v2f a0 = *(const v2f*)(arow);
        v2f a1 = *(const v2f*)(arow + 4);
        v8f c = {};
        c = __builtin_amdgcn_wmma_f32_16x16x4_f32(false, a0, false, b0, (short)0, c, false, false);
        c = __builtin_amdgcn_wmma_f32_16x16x4_f32(false, a1, false, b1, (short)0, c, false, false);

        float sc[8];
        float tm = -3.0e38f;
#pragma unroll
        for (int j = 0; j < 8; ++j) { sc[j] = c[j] * inv_sqrt_k; tm = fmaxf(tm, sc[j]); }
        tm = fmaxf(tm, __shfl_xor(tm, 16, 32));            // column max across lane pair
        float mn   = fmaxf(m, tm);
        float corr = __expf(m - mn);
        ssum *= corr;
        vacc *= corr;
        const int rbase = l0 + half * 8;
#pragma unroll
        for (int j = 0; j < 8; ++j) {
            float p = __expf(sc[j] - mn);
            ssum += p;
            vacc += p * xs[rbase + j];
        }
        m = mn;
    }
    float stot = ssum + __shfl_xor(ssum, 16, 32);
    float vtot = vacc + __shfl_xor(vacc, 16, 32);
    if (half == 0) V[(size_t)bd * O_ + o] = vtot / stot;
}

// ---------------------------------------------------------------------------
// Kernel 5: BatchNorm batch statistics per feature d over (B, O).
// ---------------------------------------------------------------------------
__global__ void __launch_bounds__(256)
bn_stats_kernel(const float* __restrict__ V, float* __restrict__ mu,
                float* __restrict__ rstd) {
    __shared__ float ssum[256], ssq[256];
    const int d = blockIdx.x;
    float s = 0.f, q = 0.f;
    for (int i = threadIdx.x; i < B_ * O_; i += 256) {
        int b = i / O_, o = i % O_;
        float v = V[((size_t)(b * D_) + d) * O_ + o];
        s += v; q += v * v;
    }
    ssum[threadIdx.x] = s; ssq[threadIdx.x] = q;
    __syncthreads();
    for (int st = 128; st > 0; st >>= 1) {
        if (threadIdx.x < st) {
            ssum[threadIdx.x] += ssum[threadIdx.x + st];
            ssq[threadIdx.x]  += ssq[threadIdx.x + st];
        }
        __syncthreads();
    }
    if (threadIdx.x == 0) {
        float n = (float)(B_ * O_);
        float mean = ssum[0] / n;
        float var  = ssq[0] / n - mean * mean;
        mu[d]   = mean;
        rstd[d] = rsqrtf(var + 1e-5f);
    }
}

// ---------------------------------------------------------------------------
// TDM: issue a 2D tensor_load_to_lds moving 16 rows x 512 cols of f32 (32KB)
// of w1 (row-major, stride 512) into LDS at byte offset lds_off.
// D# per CDNA5 ISA ch.8: group0 = {count, lds_addr, global_addr, type=2},
// group1 = {data_size=4B, tensor dims/strides, tile dims}.
// ---------------------------------------------------------------------------
#define TDM_ROWS 16

__device__ __forceinline__ void tdm_load_w1_chunk(const float* w1, int l0,
                                                  unsigned lds_off) {
    unsigned long long ga =
        (unsigned long long)(uintptr_t)(w1 + (size_t)l0 * DM_);
    u32x4 g0;
    g0[0] = 1u;                                        // count=1 (user, load)
    g0[1] = lds_off;                                   // lds_addr (bytes)
    g0[2] = (unsigned)(ga & 0xFFFFFFFFu);              // global_addr[31:0]
    g0[3] = (unsigned)((ga >> 32) & 0x01FFFFFFu)       // global_addr[56:32]
          | (2u << 30);                                // type = 2 (image)
    u32x8 g1;
    g1[0] = (2u << 16);                                // data_size = 4 bytes
    g1[1] = ((unsigned)DM_ & 0xFFFFu) << 16;           // tensor_dim0[15:0]
    g1[2] = ((unsigned)DM_ >> 16)                      // tensor_dim0[31:16]
          | (((unsigned)L_ & 0xFFFFu) << 16);          // tensor_dim1[15:0]
    g1[3] = ((unsigned)L_ >> 16)                       // tensor_dim1[31:16]
          | ((unsigned)DM_ << 16);                     // tile_dim0 = 512
    g1[4] = (unsigned)TDM_ROWS;                        // tile_dim1 = 16 rows
    g1[5] = (unsigned)DM_;                             // tensor_dim0_stride lo
    g1[6] = 0u;
    g1[7] = 0u;
    asm volatile("tensor_load_to_lds %0, %1" :: "s"(g0), "s"(g1) : "memory");
}

// ---------------------------------------------------------------------------
// Kernel 6: per-batch GEMM h = gelu(error^T(32x512) @ w1(512x512) + b1) via
// f32 WMMA 16x16x4.  w1 K-chunks (16 rows = 32KB) are staged into LDS by the
// Tensor Data Mover, double-buffered so DMA of chunk c+1 overlaps WMMA on
// chunk c.  Wave0 issues/waits TDM (TENSORcnt); one barrier/chunk publishes.
// ---------------------------------------------------------------------------
__global__ void __launch_bounds__(256)
mlp_gemm_kernel(const float* __restrict__ err, const float* __restrict__ w1,
                const float* __restrict__ b1, float* __restrict__ h) {
    __shared__ float w1s[2][TDM_ROWS * DM_];           // 2 x 32KB
    const int b    = blockIdx.x;
    const int wid  = threadIdx.x >> 5;
    const int lane = threadIdx.x & 31;
    const int half = lane >> 4;
    const int lr   = lane & 15;
    const int m_tile  = wid & 1;   // D rows 0-15 / 16-31
    const int n_group = wid >> 1;  // 0..3, each covers 8 N-tiles (128 cols)

    const unsigned lds0 = (unsigned)(uintptr_t)(&w1s[0][0]);
    const unsigned lds1 = (unsigned)(uintptr_t)(&w1s[1][0]);

    const float* arow = err + ((size_t)(b * D_) + m_tile * 16 + lr) * L_ + half * 2;
    v8f acc[8] = {};

    const int NCH = L_ / TDM_ROWS;                     // 32 chunks
    if (wid == 0) tdm_load_w1_chunk(w1, 0, lds0);      // prologue: chunk 0
    for (int c = 0; c < NCH; ++c) {
        const int buf = c & 1;
        if (wid == 0) __builtin_amdgcn_s_wait_tensorcnt(0);  // chunk c landed
        __syncthreads();                               // publish buf; other buf free
        if (wid == 0 && c + 1 < NCH)
            tdm_load_w1_chunk(w1, (c + 1) * TDM_ROWS, (c & 1) ? lds0 : lds1);

        const float* wsb = &w1s[buf][0];
#pragma unroll
        for (int kk = 0; kk < TDM_ROWS / 4; ++kk) {
            const int l0g = c * TDM_ROWS + kk * 4;     // global k (L) base
            v2f a = *(const v2f*)(arow + l0g);         // A[m=lr, k=l0g+{0,1}/{2,3}]
            const int lAl = kk * 4 + half * 2;         // local row in LDS chunk
#pragma unroll
            for (int j = 0; j < 8; ++j) {
                int n = (n_group * 8 + j) * 16 + lr;
                v2f bb;
                bb[0] = wsb[(size_t)lAl * DM_ + n];        // B[k, n]
                bb[1] = wsb[(size_t)(lAl + 1) * DM_ + n];  // B[k+1, n]
                acc[j] = __builtin_amdgcn_wmma_f32_16x16x4_f32(
                    false, a, false, bb, (short)0, acc[j], false, false);
            }
        }
        __syncthreads();                               // all reads of buf done
    }

    // epilogue: bias + exact GELU, write h[b][d][n]
#pragma unroll
    for (int j = 0; j < 8; ++j) {
        int n = (n_group * 8 + j) * 16 + lr;
        float bias = b1[n];
#pragma unroll
        for (int v = 0; v < 8; ++v) {
            int drow = m_tile * 16 + v + half * 8;
            float z = acc[j][v] + bias;
            float g = 0.5f * z * (1.f + erff(z * 0.7071067811865475f));
            h[((size_t)(b * D_) + drow) * DM_ + n] = g;
        }
    }
}

// ---------------------------------------------------------------------------
// Kernel 7: gating weights w = softmax(h @ w2 + b2) over 2, one wave per (b,d).
// ---------------------------------------------------------------------------
__global__ void __launch_bounds__(256)
gate_kernel(const float* __restrict__ h, const float* __restrict__ w2,
            const float* __restrict__ b2, float* __restrict__ wgate) {
    const int g = blockIdx.x * 8 + (threadIdx.x >> 5);  // b*D + d
    const int lane = threadIdx.x & 31;
    const float* hr = h + (size_t)g * DM_;
    float p0 = 0.f, p1 = 0.f;
    for (int mi = lane; mi < DM_; mi += 32) {
        float hv = hr[mi];
        p0 += hv * w2[mi * 2 + 0];
        p1 += hv * w2[mi * 2 + 1];
    }
#pragma unroll
    for (int off = 16; off > 0; off >>= 1) {
        p0 += __shfl_xor(p0, off, 32);
        p1 += __shfl_xor(p1, off, 32);
    }
    if (lane == 0) {
        p0 += b2[0]; p1 += b2[1];
        float mx = fmaxf(p0, p1);
        float e0 = __expf(p0 - mx), e1 = __expf(p1 - mx);
        float inv = 1.f / (e0 + e1);
        wgate[(size_t)g * 2 + 0] = e0 * inv;
        wgate[(size_t)g * 2 + 1] = e1 * inv;
    }
}

// ---------------------------------------------------------------------------
// Kernel 8: pred[b,o,d] = ymean*w0 + batchnorm(V)*w1
// ---------------------------------------------------------------------------
__global__ void final_kernel(const float* __restrict__ ymean, const float* __restrict__ V,
                             const float* __restrict__ mu, const float* __restrict__ rstd,
                             const float* __restrict__ gamma, const float* __restrict__ beta,
                             const float* __restrict__ wgate, float* __restrict__ out) {
    int idx = blockIdx.x * blockDim.x + threadIdx.x;
    if (idx >= B_ * O_ * D_) return;
    int d = idx % D_;
    int o = (idx / D_) % O_;
    int b = idx / (O_ * D_);
    int bd = b * D_ + d;
    float yv = (V[(size_t)bd * O_ + o] - mu[d]) * rstd[d] * gamma[d] + beta[d];
    out[idx] = ymean[idx] * wgate[bd * 2 + 0] + yv * wgate[bd * 2 + 1];
}

// ---------------------------------------------------------------------------
extern "C" void kernel_launch(void* const* d_in, const int* in_sizes, int n_in,
                              void* d_out, int out_size, void* d_ws, size_t ws_size,
                              hipStream_t stream) {
    const float* x      = (const float*)d_in[0];
    const float* x_date = (const float*)d_in[1];
    const float* y_date = (const float*)d_in[2];
    const float* w1     = (const float*)d_in[3];
    const float* b1     = (const float*)d_in[4];
    const float* w2     = (const float*)d_in[5];
    const float* b2     = (const float*)d_in[6];
    const float* gamma  = (const float*)d_in[7];
    const float* beta   = (const float*)d_in[8];
    float* out = (float*)d_out;

    float* ws = (float*)d_ws;
    size_t off = 0;
    float* med_t = ws + off; off += B_ * D_;
    float* std_t = ws + off; off += B_ * D_;
    float* med_m = ws + off; off += B_ * D_ * K_;
    float* std_m = ws + off; off += B_ * D_ * K_;
    float* xmap  = ws + off; off += (size_t)B_ * D_ * L_ * KP_;
    float* err   = ws + off; off += (size_t)B_ * D_ * L_;
    float* ymap  = ws + off; off += (size_t)B_ * D_ * O_ * KP_;
    float* ymean = ws + off; off += (size_t)B_ * O_ * D_;
    float* Vb    = ws + off; off += (size_t)B_ * D_ * O_;
    float* mu    = ws + off; off += D_;
    float* rstd  = ws + off; off += D_;
    float* h     = ws + off; off += (size_t)B_ * D_ * DM_;
    float* wg    = ws + off; off += (size_t)B_ * D_ * 2;

    sort_stats_kernel<<<B_ * D_,       L_, 0, stream>>>(x,      D_,      med_t, std_t);
    sort_stats_kernel<<<B_ * D_ * K_,  L_, 0, stream>>>(x_date, D_ * K_, med_m, std_m);

    map_x_kernel<<<(B_ * L_ * D_ + 255) / 256, 256, 0, stream>>>(
        x, x_date, med_t, std_t, med_m, std_m, xmap, err);
    map_y_kernel<<<(B_ * O_ * D_ + 255) / 256, 256, 0, stream>>>(
        y_date, med_t, std_t, med_m, std_m, ymap, ymean);

    attn_kernel<<<B_ * D_, 384, 0, stream>>>(x, xmap, ymap, Vb);
    bn_stats_kernel<<<D_, 256, 0, stream>>>(Vb, mu, rstd);

    mlp_gemm_kernel<<<B_, 256, 0, stream>>>(err, w1, b1, h);
    gate_kernel<<<(B_ * D_) / 8, 256, 0, stream>>>(h, w2, b2, wg);

    final_kernel<<<(B_ * O_ * D_ + 255) / 256, 256, 0, stream>>>(
        ymean, Vb, mu, rstd, gamma, beta, wg, out);
}